// SPGG_Qlearning_14508399526687
// MI455X (gfx1250) — compile-verified
//
#include <hip/hip_runtime.h>
#include <cstdint>

#ifndef __has_builtin
#define __has_builtin(x) 0
#endif

#if __has_builtin(__builtin_amdgcn_global_load_async_to_lds_b32)
#define HAVE_ASYNC_LDS 1
#else
#define HAVE_ASYNC_LDS 0
#endif

#define LSIDE  2048
#define LMASK  2047
#define LSHIFT 11
#define TILE   32
#define HALO   2
#define HT     (TILE + 2 * HALO)   /* 36: halo tile side        */
#define BT     (TILE + 2)          /* 34: base (radius-1) side  */
#define NCELL  (LSIDE * LSIDE)

typedef __attribute__((address_space(1))) int gint_t;
typedef __attribute__((address_space(3))) int lint_t;

// Match JAX plus_sum add order: ((((x + north) + south) + west) + east)
__device__ __forceinline__ float plus5(float c, float n, float s, float w, float e) {
    return (((c + n) + s) + w) + e;
}

// ---------------------------------------------------------------------------
// Stage 1: profit (double plus-stencil) + Q-learning update + pm (=upd)
// ---------------------------------------------------------------------------
__global__ __launch_bounds__(256) void spgg_stage1(
    const int*   __restrict__ type_tm,   // type_t_minus [L*L]
    const int*   __restrict__ type_t,    // type_t       [L*L]
    const float* __restrict__ Q,         // Q_tensor     [N,4]
    float*       __restrict__ Qnew,      // out: Q_new   [N,4]
    float*       __restrict__ profit_out,// out: profit  [N]
    float*       __restrict__ pm_out)    // ws : pm      [N]
{
    __shared__ int   s_t[HT * HT];   // type halo tile (values are 0/1 -> c mask)
    __shared__ float s_b[BT * BT];   // base = coop_num * R/5 on radius-1 region

    const int tid = threadIdx.x;
    const int bx  = blockIdx.x * TILE;
    const int by  = blockIdx.y * TILE;

    // ---- halo tile of type_t -> LDS (async-to-LDS on CDNA5 when available) ----
    for (int i = tid; i < HT * HT; i += 256) {
        const int ly = i / HT;
        const int lx = i - ly * HT;
        const int gy = (by + ly - HALO) & LMASK;
        const int gx = (bx + lx - HALO) & LMASK;
        const int* src = type_t + ((gy << LSHIFT) | gx);
#if HAVE_ASYNC_LDS
        __builtin_amdgcn_global_load_async_to_lds_b32(
            (gint_t*)src, (lint_t*)&s_t[i], /*offset=*/0, /*cpol=*/0);
#else
        s_t[i] = *src;
#endif
    }
#if HAVE_ASYNC_LDS
#if __has_builtin(__builtin_amdgcn_s_wait_asynccnt)
    __builtin_amdgcn_s_wait_asynccnt(0);
#else
    asm volatile("s_wait_asynccnt 0x0" ::: "memory");
#endif
#endif
    __syncthreads();

    // ---- base = plus_sum(c) * (R/5) over the 34x34 inner region ----
    const float R_OVER_5 = (float)(2.777 / 5.0);
    for (int i = tid; i < BT * BT; i += 256) {
        const int ly = i / BT;                    // 0..33 -> halo coord ly+1
        const int lx = i - ly * BT;
        const int hc = (ly + 1) * HT + (lx + 1);
        // integer 5-point sum (exact), order irrelevant for ints
        const int coop = s_t[hc] + s_t[hc - HT] + s_t[hc + HT]
                       + s_t[hc - 1] + s_t[hc + 1];
        s_b[i] = (float)coop * R_OVER_5;
    }
    __syncthreads();

    // ---- per-cell: profit + Q update; 4 cells per thread (rows ty, ty+8, ...) ----
    const int tx  = tid & 31;   // lane = x within tile -> coalesced along x
    const int ty0 = tid >> 5;   // 0..7
#pragma unroll
    for (int k = 0; k < 4; ++k) {
        const int ry = ty0 + 8 * k;                 // row in tile
        const int bi = (ry + 1) * BT + (tx + 1);    // base-coord center
        const float bc = s_b[bi];
        const float bn = s_b[bi - BT];
        const float bs = s_b[bi + BT];
        const float bw = s_b[bi - 1];
        const float be = s_b[bi + 1];
        const float ps  = plus5(bc, bn, bs, bw, be);                               // defector payoff
        const float ps1 = plus5(bc - 1.f, bn - 1.f, bs - 1.f, bw - 1.f, be - 1.f); // cooperator payoff
        const int   ctr = s_t[(ry + HALO) * HT + (tx + HALO)];                     // B = type_t (0/1)
        const float profit = ctr ? ps1 : ps;

        const int gy = by + ry;
        const int gx = bx + tx;
        const int g  = (gy << LSHIFT) | gx;

        const int A = type_tm[g];
        const int B = ctr;

        const float4 q = ((const float4*)Q)[g];     // [Q00, Q01, Q10, Q11]
        const float qb0 = B ? q.z : q.x;            // Q[C, B, 0]
        const float qb1 = B ? q.w : q.y;            // Q[C, B, 1]
        const float mv  = fmaxf(qb0, qb1);
        const float target = profit + 0.8f * mv;    // profit + GAMMA*max

        const int idx = (A << 1) | B;               // flattened (A,B)
        const float old = (idx == 0) ? q.x : (idx == 1) ? q.y : (idx == 2) ? q.z : q.w;
        const float upd = 0.2f * old + 0.8f * target;   // (1-ETA)*old + ETA*target

        float4 qo;
        qo.x = (idx == 0) ? upd : q.x;
        qo.y = (idx == 1) ? upd : q.y;
        qo.z = (idx == 2) ? upd : q.z;
        qo.w = (idx == 3) ? upd : q.w;
        ((float4*)Qnew)[g] = qo;

        profit_out[g] = profit;
        pm_out[g]     = upd;     // fermi profit matrix
    }
}

// ---------------------------------------------------------------------------
// Stage 2: Fermi imitation update (needs globally complete pm)
// ---------------------------------------------------------------------------
__global__ __launch_bounds__(256) void spgg_fermi(
    const int*   __restrict__ type_t,
    const int*   __restrict__ ldir,
    const float* __restrict__ lprob,
    const float* __restrict__ pm,
    int*         __restrict__ type_out)
{
    const int g = blockIdx.x * 256 + threadIdx.x;
    const int y = g >> LSHIFT;
    const int x = g & LMASK;

    const int d = ldir[g] & 3;
    // dir: 0 -> left (x-1), 1 -> right (x+1), 2 -> up (y-1), 3 -> down (y+1)
    const int  horiz = (d < 2) ? 1 : 0;
    const int  delta = (d & 1) ? 1 : -1;
    const int  nx = horiz ? ((x + delta) & LMASK) : x;
    const int  ny = horiz ? y : ((y + delta) & LMASK);
    const int  ng = (ny << LSHIFT) | nx;

    const float w  = 1.0f / (1.0f + expf((pm[g] - pm[ng]) * 2.0f));  // /K_FERMI, K=0.5
    const int   nb = type_t[ng];
    const int   own = type_t[g];
    type_out[g] = (lprob[g] <= w) ? nb : own;
}

// ---------------------------------------------------------------------------
// Launch
// ---------------------------------------------------------------------------
extern "C" void kernel_launch(void* const* d_in, const int* in_sizes, int n_in,
                              void* d_out, int out_size, void* d_ws, size_t ws_size,
                              hipStream_t stream)
{
    const int*   type_tm = (const int*)  d_in[0];   // type_t_minus
    const int*   type_t  = (const int*)  d_in[1];   // type_t
    const float* Q       = (const float*)d_in[2];   // Q_tensor [N,2,2]
    const int*   ldir    = (const int*)  d_in[3];   // learning_direction
    const float* lprob   = (const float*)d_in[4];   // learning_probabilities

    float* out  = (float*)d_out;
    float* Qnew = out;                                    // [0, 4N)
    int*   tout = (int*)  (out + 4 * (size_t)NCELL);      // [4N, 5N)  raw int32 bits
    float* pout = out + 5 * (size_t)NCELL;                // [5N, 6N)
    float* pm   = (float*)d_ws;                           // N floats scratch

    dim3 grid1(LSIDE / TILE, LSIDE / TILE);
    spgg_stage1<<<grid1, 256, 0, stream>>>(type_tm, type_t, Q, Qnew, pout, pm);
    spgg_fermi<<<NCELL / 256, 256, 0, stream>>>(type_t, ldir, lprob, pm, tout);
}